// PainnUpdate_38001870635383
// MI455X (gfx1250) — compile-verified
//
#include <hip/hip_runtime.h>
#include <stdint.h>

// ---------------------------------------------------------------------------
// PaiNN update on MI455X (gfx1250): FP32 WMMA (V_WMMA_F32_16X16X4_F32) path.
// 1 block (4 x wave32) = one 16-node tile. TDM stages the contiguous 16x480
// x_spherical tile into LDS; all matmuls are 16x16xK WMMA tile GEMMs.
// Wave-uniform work distribution via readfirstlane -> scalar loop control;
// contiguous A-fragments use ds_load_b64 (v2f LDS loads).
// ---------------------------------------------------------------------------

typedef __attribute__((ext_vector_type(2))) float    v2f;
typedef __attribute__((ext_vector_type(8))) float    v8f;
typedef __attribute__((ext_vector_type(4))) uint32_t v4u;
typedef __attribute__((ext_vector_type(8))) int      v8i;
typedef __attribute__((ext_vector_type(4))) int      v4i;

typedef __attribute__((address_space(1))) const float gcf;  // global const f32
typedef __attribute__((address_space(1))) float       gf;   // global f32

#define N_NODES 100000
#define TILE    16
#define SPH     480
#define SCA     128
#define NIR     224
#define MLPIN   352
#define HID     480

// LDS float offsets (dynamic shared: 23040 floats = 92160 bytes)
//  region A (7680 f): xs tile 16x480 -> mlp_in 16x352 + h 16x128 -> eq 16x224 + inner 16x128
//  region B (7680 f): U_sph 16x480
//  region C (7680 f): V_sph 16x480
#define OFF_A  0
#define OFF_U  7680
#define OFF_V  15360
#define LDS_FLOATS 23040

union Acc { v8f v; float f[8]; };

__device__ __forceinline__ v8f wmma4(v2f a, v2f b, v8f c) {
  // 8-arg form: (neg_a, A, neg_b, B, c_mod, C, reuse_a, reuse_b)
  return __builtin_amdgcn_wmma_f32_16x16x4_f32(false, a, false, b, (short)0, c,
                                               false, false);
}

__global__ void __launch_bounds__(128, 1)
painn_update_wmma(const float* __restrict__ xs_g,   // [N,128]
                  const float* __restrict__ xsph_g, // [N,480]
                  const float* __restrict__ Uw0, const float* __restrict__ Uw1,
                  const float* __restrict__ Uw2, const float* __restrict__ Ub,
                  const float* __restrict__ Vw0, const float* __restrict__ Vw1,
                  const float* __restrict__ Vw2, const float* __restrict__ Vb,
                  const float* __restrict__ dotw, // [224,128]
                  const float* __restrict__ w1,   // [352,128]
                  const float* __restrict__ b1,   // [128]
                  const float* __restrict__ w2,   // [128,480]
                  const float* __restrict__ b2,   // [480]
                  float* __restrict__ out)        // [N,128] ++ [N,480]
{
  extern __shared__ float lds[];
  float* xsT   = lds + OFF_A;            // 16x480 (phase 0-1)
  float* Us    = lds + OFF_U;            // 16x480
  float* Vs    = lds + OFF_V;            // 16x480
  float* mlpin = lds + OFF_A;            // 16x352 (phase 2-3)
  float* hbuf  = lds + OFF_A + 5632;     // 16x128 (phase 3-5)
  float* eq    = lds + OFF_A;            // 16x224 (phase 4)
  float* innr  = lds + OFF_A + 3584;     // 16x128 (phase 4b-5)

  gcf* xsG   = (gcf*)xs_g;
  gcf* xsphG = (gcf*)xsph_g;

  const int tid   = threadIdx.x;
  const int lane  = tid & 31;
  // Wave-uniform wave ID: keeps loop bounds/pointers in SGPRs.
  const int wid   = __builtin_amdgcn_readfirstlane(tid >> 5);
  const int node0 = blockIdx.x * TILE;

  const int arow  = lane & 15;          // M row within tile
  const int khalf = (lane >> 4) << 1;   // 0 or 2: K sub-offset of half-wave
  const int rbase = (lane >> 4) << 3;   // 0 or 8: M offset of C rows

  const v8f vzero = {0.f, 0.f, 0.f, 0.f, 0.f, 0.f, 0.f, 0.f};

  // ---- Phase 0: TDM DMA of the contiguous x_spherical tile into LDS -------
  if (wid == 0) {
    uint64_t ga     = (uint64_t)(uintptr_t)(xsph_g + (size_t)node0 * SPH);
    uint32_t ldsoff = __builtin_amdgcn_groupstaticsize() + OFF_A * 4u;
    v4u g0;
    g0[0] = 1u;                                              // count=1, user D#
    g0[1] = ldsoff;                                          // lds_addr (bytes)
    g0[2] = (uint32_t)ga;                                    // global_addr lo
    g0[3] = (uint32_t)((ga >> 32) & 0x1FFFFFFu) | (2u << 30);// addr hi | type=2
    v8i g1;
    g1[0] = (int)(2u << 16);                 // data_size = 4B
    g1[1] = (int)((7680u & 0xFFFFu) << 16);  // tensor_dim0 = 7680 (lo16)
    g1[2] = (int)(1u << 16);                 // tensor_dim0 hi | tensor_dim1 = 1
    g1[3] = (int)((7680u & 0xFFFFu) << 16);  // tile_dim0 = 7680 (1-D copy)
    g1[4] = 0;                               // tile_dim1 = tile_dim2 = 0
    g1[5] = 7680;                            // tensor_dim0_stride
    g1[6] = 0;
    g1[7] = 0;
    v4i gz4 = {0, 0, 0, 0};
    v8i gz8 = {0, 0, 0, 0, 0, 0, 0, 0};
    __builtin_amdgcn_tensor_load_to_lds(g0, g1, gz4, gz4, gz8, 0);
    __builtin_amdgcn_s_wait_tensorcnt(0);
  }
  __syncthreads();

  // ---- Phase 1: U_sph / V_sph irrep linears (60 x 16x16 WMMA tiles) -------
  for (int t = wid; t < 60; t += 4) {
    const int sel = t / 30;
    const int id  = t % 30;
    int l, dd, nt;
    if (id < 8)       { l = 0; dd = 0;              nt = id; }
    else if (id < 20) { l = 1; dd = (id - 8) >> 2;  nt = (id - 8) & 3; }
    else              { l = 2; dd = (id - 20) >> 1; nt = (id - 20) & 1; }
    const int   K    = (l == 0) ? 128 : (l == 1 ? 64 : 32);
    const int   xoff = (l == 0) ? 0   : (l == 1 ? 128 : 320);
    const int   es   = (l == 0) ? 1   : (l == 1 ? 3 : 5);
    const float sc   = (l == 0) ? 0.08838834764831845f
                                : (l == 1 ? 0.125f : 0.1767766952966369f);
    const float* W = sel ? (l == 0 ? Vw0 : (l == 1 ? Vw1 : Vw2))
                         : (l == 0 ? Uw0 : (l == 1 ? Uw1 : Uw2));
    const float* Bv = sel ? Vb : Ub;
    float*       O  = sel ? Vs : Us;

    int  aoff = arow * SPH + xoff + dd + khalf * es;
    gcf* Bp   = (gcf*)W + (khalf * K) + (nt * 16 + (lane & 15));

    Acc acc; acc.v = vzero;
    if (es == 1) {  // l=0: contiguous A pair -> ds_load_b64 (uniform branch)
#pragma unroll 4
      for (int ks = 0; ks < K; ks += 4) {
        v2f a = *(const v2f*)(xsT + aoff);
        v2f b; b.x = Bp[0]; b.y = Bp[K];
        acc.v = wmma4(a, b, acc.v);
        aoff += 4;
        Bp   += 4 * K;
      }
    } else {        // l=1/2: component-strided A
#pragma unroll 4
      for (int ks = 0; ks < K; ks += 4) {
        v2f a; a.x = xsT[aoff]; a.y = xsT[aoff + es];
        v2f b; b.x = Bp[0];     b.y = Bp[K];
        acc.v = wmma4(a, b, acc.v);
        aoff += 4 * es;
        Bp   += 4 * K;
      }
    }
    const int   col  = nt * 16 + (lane & 15);
    const float bias = (l == 0) ? ((gcf*)Bv)[col] : 0.0f;
    const int   oel  = xoff + col * es + dd;
#pragma unroll
    for (int r = 0; r < 8; ++r)
      O[(r + rbase) * SPH + oel] = acc.f[r] * sc + bias;
  }
  __syncthreads();

  // ---- Phase 2: mlp_in = [x_scalar | invariant(V_sph)] ---------------------
  for (int idx = tid; idx < TILE * SCA; idx += 128) {
    int n = idx >> 7, c = idx & 127;
    mlpin[n * MLPIN + c] = xsG[(size_t)(node0 + n) * SCA + c];
  }
  for (int idx = tid; idx < TILE * NIR; idx += 128) {
    int n = idx / NIR, ir = idx - n * NIR;
    const float* v = Vs + n * SPH;
    float s;
    if (ir < 128) { float x = v[ir]; s = x * x; }
    else if (ir < 192) {
      const float* p = v + 128 + (ir - 128) * 3;
      s = p[0]*p[0] + p[1]*p[1] + p[2]*p[2];
    } else {
      const float* p = v + 320 + (ir - 192) * 5;
      s = p[0]*p[0] + p[1]*p[1] + p[2]*p[2] + p[3]*p[3] + p[4]*p[4];
    }
    mlpin[n * MLPIN + SCA + ir] = sqrtf(s + 1e-12f) - 1e-6f;
  }
  __syncthreads();

  // ---- Phase 3: h = silu(mlp_in @ w1 + b1) --------------------------------
  for (int t = wid; t < 8; t += 4) {
    int  aoff = arow * MLPIN + khalf;
    gcf* Bp   = (gcf*)w1 + (khalf * SCA) + (t * 16 + (lane & 15));
    Acc acc; acc.v = vzero;
#pragma unroll 4
    for (int ks = 0; ks < MLPIN; ks += 4) {
      v2f a = *(const v2f*)(mlpin + aoff);
      v2f b; b.x = Bp[0]; b.y = Bp[SCA];
      acc.v = wmma4(a, b, acc.v);
      aoff += 4;
      Bp   += 4 * SCA;
    }
    const int   col = t * 16 + (lane & 15);
    const float bb  = ((gcf*)b1)[col];
#pragma unroll
    for (int r = 0; r < 8; ++r) {
      float z = acc.f[r] + bb;
      hbuf[(r + rbase) * SCA + col] = z / (1.0f + __expf(-z));
    }
  }
  __syncthreads();

  // ---- Phase 4a: equidot(U_sph, V_sph) ------------------------------------
  for (int idx = tid; idx < TILE * NIR; idx += 128) {
    int n = idx / NIR, ir = idx - n * NIR;
    const float* u = Us + n * SPH;
    const float* v = Vs + n * SPH;
    float s;
    if (ir < 128) s = u[ir] * v[ir];
    else if (ir < 192) {
      int q = ir - 128;
      const float* pu = u + 128 + q * 3; const float* pv = v + 128 + q * 3;
      s = (pu[0]*pv[0] + pu[1]*pv[1] + pu[2]*pv[2]) * 0.5773502691896258f;
    } else {
      int q = ir - 192;
      const float* pu = u + 320 + q * 5; const float* pv = v + 320 + q * 5;
      s = (pu[0]*pv[0] + pu[1]*pv[1] + pu[2]*pv[2] + pu[3]*pv[3] + pu[4]*pv[4])
          * 0.4472135954999579f;
    }
    eq[n * NIR + ir] = s;
  }
  __syncthreads();

  // ---- Phase 4b: inner = equidot @ dot_w ----------------------------------
  for (int t = wid; t < 8; t += 4) {
    int  aoff = arow * NIR + khalf;
    gcf* Bp   = (gcf*)dotw + (khalf * SCA) + (t * 16 + (lane & 15));
    Acc acc; acc.v = vzero;
#pragma unroll 4
    for (int ks = 0; ks < NIR; ks += 4) {
      v2f a = *(const v2f*)(eq + aoff);
      v2f b; b.x = Bp[0]; b.y = Bp[SCA];
      acc.v = wmma4(a, b, acc.v);
      aoff += 4;
      Bp   += 4 * SCA;
    }
    const int col = t * 16 + (lane & 15);
#pragma unroll
    for (int r = 0; r < 8; ++r)
      innr[(r + rbase) * SCA + col] = acc.f[r];
  }
  __syncthreads();

  // ---- Phase 5: mlp_out tiles (h @ w2 + b2), applied on the fly -----------
  gf* out_sc = (gf*)out;                            // [N,128]
  gf* out_sp = (gf*)(out + (size_t)N_NODES * SCA);  // [N,480]
  for (int t = wid; t < 22; t += 4) {
    if (t < 14) {
      // a_vv tile -> rsh_conv(U_sph) + residual into out_spherical
      const int colbase = t * 16;
      int  aoff = arow * SCA + khalf;
      gcf* Bp   = (gcf*)w2 + (khalf * HID) + (colbase + (lane & 15));
      Acc acc; acc.v = vzero;
#pragma unroll 4
      for (int ks = 0; ks < SCA; ks += 4) {
        v2f a = *(const v2f*)(hbuf + aoff);
        v2f b; b.x = Bp[0]; b.y = Bp[HID];
        acc.v = wmma4(a, b, acc.v);
        aoff += 4;
        Bp   += 4 * HID;
      }
      const int   col = colbase + (lane & 15);  // irrep index 0..223
      const float bb  = ((gcf*)b2)[col];
      int ofs, d;
      if (col < 128)      { ofs = col;                   d = 1; }
      else if (col < 192) { ofs = 128 + (col - 128) * 3; d = 3; }
      else                { ofs = 320 + (col - 192) * 5; d = 5; }
#pragma unroll
      for (int r = 0; r < 8; ++r) {
        int    nrow = r + rbase;
        float  av   = acc.f[r] + bb;
        size_t gro  = (size_t)(node0 + nrow) * SPH + ofs;
        const float* us = Us + nrow * SPH + ofs;
        for (int j = 0; j < d; ++j)
          __builtin_nontemporal_store(xsphG[gro + j] + av * us[j],
                                      &out_sp[gro + j]);
      }
    } else {
      // a_sv / a_ss tile pair -> d_scalar + residual into out_scalar
      const int j     = t - 14;
      const int cb_sv = NIR + j * 16;         // 224 + 16j
      const int cb_ss = NIR + SCA + j * 16;   // 352 + 16j
      int  aoff = arow * SCA + khalf;
      gcf* Bsv  = (gcf*)w2 + (khalf * HID) + (cb_sv + (lane & 15));
      gcf* Bss  = (gcf*)w2 + (khalf * HID) + (cb_ss + (lane & 15));
      Acc asv, ass; asv.v = vzero; ass.v = vzero;
#pragma unroll 4
      for (int ks = 0; ks < SCA; ks += 4) {
        v2f a = *(const v2f*)(hbuf + aoff);
        v2f bv; bv.x = Bsv[0]; bv.y = Bsv[HID];
        v2f bs; bs.x = Bss[0]; bs.y = Bss[HID];
        asv.v = wmma4(a, bv, asv.v);
        ass.v = wmma4(a, bs, ass.v);
        aoff += 4;
        Bsv  += 4 * HID;
        Bss  += 4 * HID;
      }
      const int   col = j * 16 + (lane & 15);  // scalar channel 0..127
      const float bsv = ((gcf*)b2)[cb_sv + (lane & 15)];
      const float bss = ((gcf*)b2)[cb_ss + (lane & 15)];
#pragma unroll
      for (int r = 0; r < 8; ++r) {
        int    nrow = r + rbase;
        float  inn  = innr[nrow * SCA + col];
        float  ds   = (asv.f[r] + bsv) * inn + (ass.f[r] + bss);
        size_t g    = (size_t)(node0 + nrow) * SCA + col;
        __builtin_nontemporal_store(xsG[g] + ds, &out_sc[g]);
      }
    }
  }
}

extern "C" void kernel_launch(void* const* d_in, const int* in_sizes, int n_in,
                              void* d_out, int out_size, void* d_ws, size_t ws_size,
                              hipStream_t stream) {
  (void)in_sizes; (void)n_in; (void)out_size; (void)d_ws; (void)ws_size;
  const float* x_scalar = (const float*)d_in[0];
  const float* x_sph    = (const float*)d_in[1];
  const float* U_w0     = (const float*)d_in[2];
  const float* U_w1     = (const float*)d_in[3];
  const float* U_w2     = (const float*)d_in[4];
  const float* U_b      = (const float*)d_in[5];
  const float* V_w0     = (const float*)d_in[6];
  const float* V_w1     = (const float*)d_in[7];
  const float* V_w2     = (const float*)d_in[8];
  const float* V_b      = (const float*)d_in[9];
  const float* dot_w    = (const float*)d_in[10];
  const float* mlp_w1   = (const float*)d_in[11];
  const float* mlp_b1   = (const float*)d_in[12];
  const float* mlp_w2   = (const float*)d_in[13];
  const float* mlp_b2   = (const float*)d_in[14];
  float*       outp     = (float*)d_out;

  const size_t shmem = LDS_FLOATS * sizeof(float);  // 92160 B (< 320 KB/WGP)
  (void)hipFuncSetAttribute(reinterpret_cast<const void*>(painn_update_wmma),
                            hipFuncAttributeMaxDynamicSharedMemorySize,
                            (int)shmem);

  dim3 grid(N_NODES / TILE);  // 6250, exact
  dim3 block(128);            // 4 x wave32
  painn_update_wmma<<<grid, block, shmem, stream>>>(
      x_scalar, x_sph, U_w0, U_w1, U_w2, U_b, V_w0, V_w1, V_w2, V_b,
      dot_w, mlp_w1, mlp_b1, mlp_w2, mlp_b2, outp);
}